// InstanceEmbeddingLoss_51539607722
// MI455X (gfx1250) — compile-verified
//
#include <hip/hip_runtime.h>
#include <hip/hip_bf16.h>
#include <math.h>

// Problem constants (fixed by the reference setup)
#define BATCH 4
#define HW 262144          // 512*512
#define NINST 32
#define MELEM (NINST * HW) // 8,388,608 elements per sample
#define P2_TPB 256
#define P2_EPT 8
#define P2_BLOCKS 4096     // 4096*256*8 == MELEM

typedef float v2f __attribute__((ext_vector_type(2)));
typedef float v8f __attribute__((ext_vector_type(8)));

struct Stats {
    unsigned cnt[BATCH][NINST];
    float s0[BATCH][NINST], s1[BATCH][NINST], ss[BATCH][NINST];
    float c0[BATCH][NINST], c1[BATCH][NINST], ms[BATCH][NINST];
    float Gf[BATCH], invsum[BATCH], stot[BATCH], dotv[BATCH];
};

// ---------------------------------------------------------------- utilities
__global__ void k_zero(unsigned* __restrict__ p, unsigned n) {
    unsigned i = blockIdx.x * blockDim.x + threadIdx.x;
    unsigned stride = gridDim.x * blockDim.x;
    for (; i < n; i += stride) p[i] = 0u;
}

// ------------------------------------------------------- per-instance stats
__global__ void k_stats(const float* __restrict__ emb, const float* __restrict__ sig,
                        const int* __restrict__ gt, Stats* st) {
    int b = blockIdx.y;
    int base = blockIdx.x * 1024;
    int t = threadIdx.x;
    __shared__ unsigned lc[NINST];
    __shared__ float l0[NINST], l1[NINST], l2[NINST];
    if (t < NINST) { lc[t] = 0u; l0[t] = 0.f; l1[t] = 0.f; l2[t] = 0.f; }
    __syncthreads();
    const float* e0 = emb + (size_t)(b * 2 + 0) * HW;
    const float* e1 = emb + (size_t)(b * 2 + 1) * HW;
    const float* sg = sig + (size_t)b * HW;
    const int*   g  = gt  + (size_t)b * HW;
    for (int k = 0; k < 4; k++) {
        int p = base + k * 256 + t;
        int lbl = g[p];
        if (lbl > 0) {
            int n = lbl - 1;
            atomicAdd(&lc[n], 1u);
            atomicAdd(&l0[n], e0[p]);
            atomicAdd(&l1[n], e1[p]);
            atomicAdd(&l2[n], sg[p]);
        }
    }
    __syncthreads();
    if (t < NINST) {
        if (lc[t]) atomicAdd(&st->cnt[b][t], lc[t]);
        atomicAdd(&st->s0[b][t], l0[t]);
        atomicAdd(&st->s1[b][t], l1[t]);
        atomicAdd(&st->ss[b][t], l2[t]);
    }
}

__global__ void k_finalize(Stats* st) {
    int t = threadIdx.x;
    if (t < BATCH * NINST) {
        int b = t >> 5, n = t & 31;
        unsigned c = st->cnt[b][n];
        float fc = c ? (float)c : 1.0f;
        st->c0[b][n] = st->s0[b][n] / fc;
        st->c1[b][n] = st->s1[b][n] / fc;
        st->ms[b][n] = st->ss[b][n] / fc;
        if (n == 0) {
            unsigned G = 0; float inv = 0.f;
            for (int i = 0; i < NINST; i++) {
                unsigned ci = st->cnt[b][i];
                G += ci;
                inv += 1.0f / (ci ? (float)ci : 1.0f);
            }
            st->Gf[b] = (float)G;
            st->invsum[b] = inv;
        }
    }
}

// ----------------------------------------------------------- variance term
__global__ void k_var(const float* __restrict__ sig, const int* __restrict__ gt, Stats* st) {
    int b = blockIdx.y;
    int base = blockIdx.x * 1024;
    int t = threadIdx.x;
    const float* sg = sig + (size_t)b * HW;
    const int*   g  = gt  + (size_t)b * HW;
    float acc = 0.f;
    for (int k = 0; k < 4; k++) {
        int p = base + k * 256 + t;
        int lbl = g[p];
        if (lbl > 0) {
            float dv = sg[p] - st->ms[b][lbl - 1];
            acc += dv * dv;
        }
    }
    __shared__ float sd[256];
    sd[t] = acc; __syncthreads();
    for (int s = 128; s > 0; s >>= 1) { if (t < s) sd[t] += sd[t + s]; __syncthreads(); }
    if (t == 0) atomicAdd(&st->stot[b], sd[0]);
}

// ------------------------------------------------ shared per-element error
__device__ __forceinline__ float elem_err(const float* __restrict__ e0, const float* __restrict__ e1,
                                          const float* __restrict__ sg, const int* __restrict__ g,
                                          const float* __restrict__ c0, const float* __restrict__ c1,
                                          const float* __restrict__ ms, int n, int p, int& lab) {
    lab = (g[p] == n + 1) ? 1 : 0;
    float d0 = e0[p] - c0[n];
    float d1 = e1[p] - c1[n];
    float dist = sqrtf(d0 * d0 + d1 * d1);
    float r = dist / ms[n];
    float pr = expf(-0.5f * r * r);
    return lab ? (2.0f - 2.0f * pr) : (2.0f * pr);
}

// ----------------------------------------------- pass 1: bucket histograms
__global__ void k_pass1(const float* __restrict__ e0, const float* __restrict__ e1,
                        const float* __restrict__ sg, const int* __restrict__ g,
                        const float* __restrict__ c0, const float* __restrict__ c1,
                        const float* __restrict__ ms,
                        unsigned* __restrict__ X1, unsigned* __restrict__ X0, int shift) {
    int t0 = blockIdx.x * (P2_TPB * P2_EPT) + threadIdx.x;
    for (int k = 0; k < P2_EPT; k++) {
        int ei = t0 + k * P2_TPB;
        int n = ei >> 18, p = ei & (HW - 1);
        int lab;
        float e = elem_err(e0, e1, sg, g, c0, c1, ms, n, p, lab);
        unsigned key = ~__float_as_uint(e);       // e >= 0 -> ascending key == descending e
        unsigned bkt = key >> shift;
        atomicAdd(lab ? &X1[bkt] : &X0[bkt], 1u);
    }
}

// --------------------------------------------------- scans over the bins
__global__ void k_scan1(const unsigned* __restrict__ X1, const unsigned* __restrict__ X0,
                        unsigned* __restrict__ bs1, unsigned* __restrict__ bs0) {
    int t = threadIdx.x;
    size_t base = (size_t)blockIdx.x * 4096;
    unsigned a = 0, c = 0;
    for (int i = 0; i < 16; i++) {
        size_t idx = base + t + (size_t)i * 256;
        a += X1[idx]; c += X0[idx];
    }
    __shared__ unsigned sa[256], sc[256];
    sa[t] = a; sc[t] = c; __syncthreads();
    for (int s = 128; s > 0; s >>= 1) { if (t < s) { sa[t] += sa[t + s]; sc[t] += sc[t + s]; } __syncthreads(); }
    if (t == 0) { bs1[blockIdx.x] = sa[0]; bs0[blockIdx.x] = sc[0]; }
}

__global__ void k_scan2(unsigned* __restrict__ bs1, unsigned* __restrict__ bs0, int n) {
    __shared__ unsigned sa[1024], sc[1024];
    int t = threadIdx.x;
    sa[t] = (t < n) ? bs1[t] : 0u;
    sc[t] = (t < n) ? bs0[t] : 0u;
    __syncthreads();
    for (int off = 1; off < 1024; off <<= 1) {
        unsigned va = (t >= off) ? sa[t - off] : 0u;
        unsigned vc = (t >= off) ? sc[t - off] : 0u;
        __syncthreads();
        sa[t] += va; sc[t] += vc;
        __syncthreads();
    }
    if (t < n) {
        bs1[t] = t ? sa[t - 1] : 0u;   // exclusive
        bs0[t] = t ? sc[t - 1] : 0u;
    }
}

// Per bucket g:  X1 <- ExOnes (ones-slot counter base), Y0 <- IncOnes (c for zeros),
//                Y1 <- ExZeros (pos = Y1 + ones-slot),  X0 <- IncOnes+ExZeros (zeros-slot base)
__global__ void k_scan3(unsigned* __restrict__ X1, unsigned* __restrict__ X0,
                        unsigned* __restrict__ Y1, unsigned* __restrict__ Y0,
                        const unsigned* __restrict__ bs1, const unsigned* __restrict__ bs0) {
    int t = threadIdx.x;
    size_t base = (size_t)blockIdx.x * 4096 + (size_t)t * 16;
    unsigned a[16], c[16], ta = 0, tc = 0;
    for (int i = 0; i < 16; i++) { a[i] = X1[base + i]; c[i] = X0[base + i]; ta += a[i]; tc += c[i]; }
    __shared__ unsigned sa[256], sc[256];
    sa[t] = ta; sc[t] = tc; __syncthreads();
    for (int off = 1; off < 256; off <<= 1) {
        unsigned va = (t >= off) ? sa[t - off] : 0u;
        unsigned vc = (t >= off) ? sc[t - off] : 0u;
        __syncthreads();
        sa[t] += va; sc[t] += vc;
        __syncthreads();
    }
    unsigned ra = bs1[blockIdx.x] + (t ? sa[t - 1] : 0u);
    unsigned rc = bs0[blockIdx.x] + (t ? sc[t - 1] : 0u);
    for (int i = 0; i < 16; i++) {
        unsigned exA = ra, incA = ra + a[i], exB = rc;
        X1[base + i] = exA;
        Y0[base + i] = incA;
        Y1[base + i] = exB;
        X0[base + i] = incA + exB;
        ra = incA; rc += c[i];
    }
}

// --------------------------------- pass 2: per-element Lovasz contribution
__global__ void k_pass2(const float* __restrict__ e0, const float* __restrict__ e1,
                        const float* __restrict__ sg, const int* __restrict__ g,
                        const float* __restrict__ c0, const float* __restrict__ c1,
                        const float* __restrict__ ms, const float* __restrict__ Gp,
                        unsigned* __restrict__ X1, unsigned* __restrict__ X0,
                        const unsigned* __restrict__ Y1, const unsigned* __restrict__ Y0,
                        int shift, float* __restrict__ partials) {
    int t0 = blockIdx.x * (P2_TPB * P2_EPT) + threadIdx.x;
    float G = *Gp;
    float acc = 0.f;
    for (int k = 0; k < P2_EPT; k++) {
        int ei = t0 + k * P2_TPB;
        int n = ei >> 18, p = ei & (HW - 1);
        int lab;
        float e = elem_err(e0, e1, sg, g, c0, c1, ms, n, p, lab);
        unsigned key = ~__float_as_uint(e);
        unsigned bkt = key >> shift;
        unsigned pos, cc;
        if (lab) {
            unsigned old = atomicAdd(&X1[bkt], 1u);  // O_g + j
            cc = old + 1u;                            // inclusive ones count
            pos = Y1[bkt] + old;                      // S_g + j
        } else {
            unsigned old = atomicAdd(&X0[bkt], 1u);  // S_g + m_g + j
            pos = old;
            cc = Y0[bkt];                             // O_g + m_g
        }
        // intersection/union are exact integers (< 2^24) -> matches reference fp32 math
        float fc = (float)cc;
        float U = G + (float)(pos + 1u) - fc;
        float Jp = 1.0f - (G - fc) / U;
        float grad;
        if (pos == 0u) {
            grad = Jp;
        } else {
            float cm = fc - (float)lab;
            float Um = G + (float)pos - cm;
            grad = Jp - (1.0f - (G - cm) / Um);
        }
        acc += fmaxf(e, 0.0f) * grad;
    }
    __shared__ float sd[P2_TPB];
    sd[threadIdx.x] = acc; __syncthreads();
    for (int s = 128; s > 0; s >>= 1) { if (threadIdx.x < s) sd[threadIdx.x] += sd[threadIdx.x + s]; __syncthreads(); }
    if (threadIdx.x == 0) partials[blockIdx.x] = sd[0];
}

// -------------- deterministic single-wave WMMA-accumulated final reduction
__global__ void k_reduce_dot(const float* __restrict__ partials, Stats* st, int smp) {
    int lane = threadIdx.x;
#if __has_builtin(__builtin_amdgcn_wmma_f32_16x16x4_f32)
    v8f acc = {0.f, 0.f, 0.f, 0.f, 0.f, 0.f, 0.f, 0.f};
    v2f ones; ones.x = 1.0f; ones.y = 1.0f;
    for (int i = 0; i < P2_BLOCKS; i += 64) {
        v2f a;
        a.x = partials[i + lane];
        a.y = partials[i + 32 + lane];
        // D[m][n] = sum_k A[m][k] * 1 + C[m][n]  -> chained accumulate of 64 values/iter
        acc = __builtin_amdgcn_wmma_f32_16x16x4_f32(false, a, false, ones,
                                                    (short)0, acc, false, false);
    }
    float s = acc[0] + acc[1] + acc[2] + acc[3] + acc[4] + acc[5] + acc[6] + acc[7];
    float o = __shfl(s, 16, 32);  // lanes 0-15 hold M=0..7, lanes 16-31 hold M=8..15
    if (lane == 0) st->dotv[smp] = s + o;
#else
    float s = 0.f;
    for (int i = lane; i < P2_BLOCKS; i += 32) s += partials[i];
    for (int off = 16; off > 0; off >>= 1) s += __shfl_down(s, off, 32);
    if (lane == 0) st->dotv[smp] = s;
#endif
}

__global__ void k_final(const Stats* __restrict__ st, float* __restrict__ out) {
    float r = 0.f;
    for (int b = 0; b < BATCH; b++)
        r += st->dotv[b] + st->stot[b] * st->invsum[b] * (1.0f / 32.0f);
    out[0] = r * (1.0f / BATCH);
}

// ---------------------------------------------------------------- launcher
extern "C" void kernel_launch(void* const* d_in, const int* in_sizes, int n_in,
                              void* d_out, int out_size, void* d_ws, size_t ws_size,
                              hipStream_t stream) {
    const float* emb = (const float*)d_in[0];
    const float* sig = (const float*)d_in[1];
    const int*   gt  = (const int*)d_in[2];
    float* out = (float*)d_out;

    // Pick the largest bucket count that fits the workspace (22 bits preferred).
    int lognb = 22;
    for (;;) {
        size_t need = (((size_t)4u << lognb) * 4u) * 4u /*X1,X0,Y1,Y0*/ +
                      2048u * 4u /*bs*/ + sizeof(Stats) + 512u /*align*/ +
                      (size_t)P2_BLOCKS * 4u;
        if (need <= ws_size || lognb <= 14) break;
        lognb -= 2;
    }
    size_t NB = (size_t)1 << lognb;
    int shift = 32 - lognb;
    int nblk = (int)(NB >> 12);  // bins per scan-block = 4096

    unsigned* w   = (unsigned*)d_ws;
    unsigned* X1  = w;
    unsigned* X0  = X1 + NB;
    unsigned* Y1  = X0 + NB;
    unsigned* Y0  = Y1 + NB;
    unsigned* bs1 = Y0 + NB;
    unsigned* bs0 = bs1 + 1024;
    Stats* st = (Stats*)(bs0 + 1024);
    float* partials = (float*)((char*)st + ((sizeof(Stats) + 255) & ~(size_t)255));

    // Stats (zeroed every call -> deterministic, no cross-call state)
    k_zero<<<1, 256, 0, stream>>>((unsigned*)st, (unsigned)(sizeof(Stats) / 4));
    k_stats<<<dim3(HW / 1024, BATCH), 256, 0, stream>>>(emb, sig, gt, st);
    k_finalize<<<1, 128, 0, stream>>>(st);
    k_var<<<dim3(HW / 1024, BATCH), 256, 0, stream>>>(sig, gt, st);

    for (int b = 0; b < BATCH; b++) {
        const float* e0 = emb + (size_t)(b * 2 + 0) * HW;
        const float* e1 = emb + (size_t)(b * 2 + 1) * HW;
        const float* sg = sig + (size_t)b * HW;
        const int*   g  = gt  + (size_t)b * HW;
        const float* c0 = st->c0[b];      // address arithmetic only (device struct)
        const float* c1 = st->c1[b];
        const float* ms = st->ms[b];
        const float* Gp = &st->Gf[b];

        k_zero<<<2048, 256, 0, stream>>>(X1, (unsigned)(2 * NB));  // X1 and X0 contiguous
        k_pass1<<<P2_BLOCKS, P2_TPB, 0, stream>>>(e0, e1, sg, g, c0, c1, ms, X1, X0, shift);
        k_scan1<<<nblk, 256, 0, stream>>>(X1, X0, bs1, bs0);
        k_scan2<<<1, 1024, 0, stream>>>(bs1, bs0, nblk);
        k_scan3<<<nblk, 256, 0, stream>>>(X1, X0, Y1, Y0, bs1, bs0);
        k_pass2<<<P2_BLOCKS, P2_TPB, 0, stream>>>(e0, e1, sg, g, c0, c1, ms, Gp,
                                                  X1, X0, Y1, Y0, shift, partials);
        k_reduce_dot<<<1, 32, 0, stream>>>(partials, st, b);
    }
    k_final<<<1, 1, 0, stream>>>(st, out);
}